// LightGCN_31499290149531
// MI455X (gfx1250) — compile-verified
//
#include <hip/hip_runtime.h>

#define N_USERS 100000
#define N_ITEMS 50000
#define N_NODES (N_USERS + N_ITEMS)
#define LATENT  64
#define NNZ     6000000
#define N_LAYERS 3
#define BATCH   2048
#define NELEM   (N_NODES * LATENT)   // 9,600,000 floats per buffer (38.4 MB)

typedef __attribute__((ext_vector_type(2))) float v2f;
typedef __attribute__((ext_vector_type(8))) float v8f;

// Single-instruction L2 atomic add (no-return -> STOREcnt). Avoids any CAS-loop
// lowering; gfx1250 supports GLOBAL_ATOMIC_ADD_F32 natively.
__device__ __forceinline__ void gatomic_add_f32(float* p, float v) {
    asm volatile("global_atomic_add_f32 %0, %1, off" : : "v"(p), "v"(v) : "memory");
}

// acc = emb = concat(user_emb, item_emb), vectorized float4.
__global__ void lg_init(const float4* __restrict__ ue, const float4* __restrict__ ie,
                        float4* __restrict__ acc, float4* __restrict__ emb) {
    int i = blockIdx.x * blockDim.x + threadIdx.x;      // float4 index
    const int nU4 = (N_USERS * LATENT) / 4;
    const int nT4 = NELEM / 4;
    if (i >= nT4) return;
    float4 v = (i < nU4) ? ue[i] : ie[i - nU4];
    acc[i] = v;
    emb[i] = v;
}

__global__ void lg_zero(float4* __restrict__ p) {
    int i = blockIdx.x * blockDim.x + threadIdx.x;
    if (i < NELEM / 4) p[i] = make_float4(0.f, 0.f, 0.f, 0.f);
}

// One nonzero per wave32: lane l handles elements [2l, 2l+1] of the 64-wide row.
// The wave index is forced wave-uniform via readfirstlane so the 12 B/nnz COO
// metadata becomes scalar (SMEM) loads instead of 32 redundant VMEM loads.
__global__ void lg_spmm(const int* __restrict__ row, const int* __restrict__ col,
                        const float* __restrict__ val,
                        const float* __restrict__ x, float* __restrict__ y) {
    int nz = __builtin_amdgcn_readfirstlane(
                 blockIdx.x * (blockDim.x >> 5) + (threadIdx.x >> 5));
    if (nz >= NNZ) return;
    int lane = threadIdx.x & 31;
    int r   = row[nz];
    int c   = col[nz];
    float v = val[nz];
    const float2* xr = (const float2*)(x + (size_t)c * LATENT);
    float2 xv = xr[lane];
    float* dst = y + (size_t)r * LATENT + lane * 2;
    gatomic_add_f32(dst,     v * xv.x);
    gatomic_add_f32(dst + 1, v * xv.y);
}

__global__ void lg_add(float4* __restrict__ acc, const float4* __restrict__ emb) {
    int i = blockIdx.x * blockDim.x + threadIdx.x;
    if (i >= NELEM / 4) return;
    float4 a = acc[i];
    float4 b = emb[i];
    a.x += b.x; a.y += b.y; a.z += b.z; a.w += b.w;
    acc[i] = a;
}

// gamma[b] = dot(light_out[users[b]], light_out[N_USERS+items[b]])
//          = (1/16) * diag( U_tile (16x64) x I_tile^T (64x16) )
// One wave per 16 pairs; 16 chained v_wmma_f32_16x16x4_f32 over K=64.
// A 16x4 f32 layout: lanes 0-15 hold K={0,1}, lanes 16-31 hold K={2,3} for row M=lane&15.
// B 4x16 mirrors it with B[k][n] = I[n][k], so both sides use identical float2 gathers.
__global__ void lg_gamma(const float* __restrict__ acc,
                         const int* __restrict__ users, const int* __restrict__ items,
                         float* __restrict__ out) {
    __shared__ float sh[256];
    int lane = threadIdx.x;          // blockDim.x == 32, EXEC all-ones (WMMA requirement)
    int m    = lane & 15;
    int hi   = lane >> 4;            // 0: K pair {0,1}, 1: K pair {2,3}
    int tile = blockIdx.x;           // 128 tiles of 16 pairs
    int ur = users[tile * 16 + m];
    int ir = items[tile * 16 + m];
    const float* U = acc + (size_t)ur * LATENT;
    const float* I = acc + (size_t)(N_USERS + ir) * LATENT;
    v8f c = {0.f, 0.f, 0.f, 0.f, 0.f, 0.f, 0.f, 0.f};
#pragma unroll
    for (int kk = 0; kk < 16; ++kk) {
        int kb = kk * 4 + hi * 2;
        v2f a = *(const v2f*)(U + kb);
        v2f b = *(const v2f*)(I + kb);
        c = __builtin_amdgcn_wmma_f32_16x16x4_f32(
                /*neg_a=*/false, a, /*neg_b=*/false, b,
                /*c_mod=*/(short)0, c, /*reuse_a=*/false, /*reuse_b=*/false);
    }
    // D layout: VGPR j -> M=j (lanes 0-15, N=lane) / M=j+8 (lanes 16-31, N=lane-16)
#pragma unroll
    for (int j = 0; j < 8; ++j) sh[(j + 8 * hi) * 16 + m] = c[j];
    __syncthreads();
    if (lane < 16) out[tile * 16 + lane] = sh[lane * 16 + lane] * 0.0625f; // (1/4)*(1/4)
}

extern "C" void kernel_launch(void* const* d_in, const int* in_sizes, int n_in,
                              void* d_out, int out_size, void* d_ws, size_t ws_size,
                              hipStream_t stream) {
    (void)in_sizes; (void)n_in; (void)out_size; (void)ws_size;
    const int*   row   = (const int*)d_in[0];
    const int*   col   = (const int*)d_in[1];
    const float* vals  = (const float*)d_in[2];
    const float* ue    = (const float*)d_in[3];
    const float* ie    = (const float*)d_in[4];
    const int*   users = (const int*)d_in[5];
    const int*   items = (const int*)d_in[6];
    float* out = (float*)d_out;

    float* acc  = (float*)d_ws;          // 38.4 MB
    float* bufA = acc  + NELEM;          // 38.4 MB
    float* bufB = bufA + NELEM;          // 38.4 MB  (total 115.2 MB, L2-resident)

    const int t4 = NELEM / 4;
    dim3 blk(256);
    dim3 gElem((t4 + 255) / 256);

    lg_init<<<gElem, blk, 0, stream>>>((const float4*)ue, (const float4*)ie,
                                       (float4*)acc, (float4*)bufA);

    float* cur = bufA;
    float* nxt = bufB;
    for (int l = 0; l < N_LAYERS; ++l) {
        lg_zero<<<gElem, blk, 0, stream>>>((float4*)nxt);
        lg_spmm<<<dim3((NNZ + 7) / 8), blk, 0, stream>>>(row, col, vals, cur, nxt);
        lg_add<<<gElem, blk, 0, stream>>>((float4*)acc, (const float4*)nxt);
        float* t = cur; cur = nxt; nxt = t;
    }

    lg_gamma<<<dim3(BATCH / 16), dim3(32), 0, stream>>>(acc, users, items, out);
}